// GINLayer_71416716197907
// MI455X (gfx1250) — compile-verified
//
#include <hip/hip_runtime.h>

typedef __attribute__((ext_vector_type(2))) float v2f;
typedef __attribute__((ext_vector_type(8))) float v8f;

static constexpr int Dd     = 128;
static constexpr int NNODES = 100000;
static constexpr int NEDGES = 1600000;
static constexpr int BROWS  = 128;                              // rows per block (8 waves x 16)
static constexpr int NBLK   = (NNODES + BROWS - 1) / BROWS;     // 782
static constexpr int BSTR2  = 65;                               // B LDS row stride in float2 (130 floats)

// ---------------- init: H = x (the (1+eps)*x self term, eps=0), zero stats ----------------
__global__ void init_kernel(const float4* __restrict__ x4, float4* __restrict__ h4,
                            float* __restrict__ stats) {
  int gid = blockIdx.x * blockDim.x + threadIdx.x;
  if (gid < NNODES * Dd / 4) h4[gid] = x4[gid];
  if (gid < 1024) stats[gid] = 0.f;
}

// ---------------- edge scatter: H[dst] += x[src] (hardware fp32 atomics into L2) ----------
__global__ void scatter_kernel(const float4* __restrict__ x4,
                               const long long* __restrict__ ei,
                               float* __restrict__ H) {
  int gid = blockIdx.x * blockDim.x + threadIdx.x;   // one thread per (edge, float4-quad)
  int e = gid >> 5;
  int q = gid & 31;
  if (e >= NEDGES) return;
  int src = (int)ei[e];
  int dst = (int)ei[NEDGES + e];
  float4 v = x4[src * 32 + q];
  float* o = H + (size_t)dst * Dd + q * 4;
  unsafeAtomicAdd(o + 0, v.x);
  unsafeAtomicAdd(o + 1, v.y);
  unsafeAtomicAdd(o + 2, v.z);
  unsafeAtomicAdd(o + 3, v.w);
}

// ---------------- fused GEMM: Y = act(A) @ W^T (+ bias), optional BN stats ----------------
// act(A) = relu(A*tf_scale + tf_shift) when DO_TF, else identity. All modes compile-time.
template <bool DO_TF, bool HAS_BIAS, bool DO_STATS, bool RELU_OUT>
__global__ __launch_bounds__(256) void gemm_bn_kernel(
    const float* __restrict__ A, const float* __restrict__ W,
    const float* __restrict__ bias,
    const float* __restrict__ tf_scale, const float* __restrict__ tf_shift,
    float* __restrict__ Y,
    float* __restrict__ gsum, float* __restrict__ gsq)
{
  extern __shared__ float smem[];        // Bs[n][k] = W[n][k], 128 rows, stride 130 floats
  __shared__ float sstat[2 * Dd];
  __shared__ float stf[2 * Dd];          // [0..127]=scale, [128..255]=shift

  const int tid  = threadIdx.x;
  const int wave = tid >> 5;
  const int lane = tid & 31;
  const int half = lane >> 4;
  const int l16  = lane & 15;
  const int m0   = blockIdx.x * BROWS;

  {
    v2f* B2w = (v2f*)smem;
    const v2f* W2 = (const v2f*)W;
    #pragma unroll
    for (int i = 0; i < (Dd * Dd / 2) / 256; ++i) {
      int idx = tid + i * 256;
      int n = idx >> 6;
      int k2 = idx & 63;
      B2w[n * BSTR2 + k2] = W2[idx];
    }
  }
  if (DO_STATS) {
    if (tid < 2 * Dd) sstat[tid] = 0.f;
  }
  if (DO_TF) {
    if (tid < Dd) {
      stf[tid]      = tf_scale[tid];
      stf[Dd + tid] = tf_shift[tid];
    }
  }
  __syncthreads();

  const int wrow = m0 + wave * 16;
  int arow = wrow + l16;                      // A-frag rows: each wave reads only its 16 rows
  if (arow > NNODES - 1) arow = NNODES - 1;   // tail clamp (masked at store/stats)
  const v2f* Arow2 = (const v2f*)(A + (size_t)arow * Dd);
  const v2f* B2  = (const v2f*)smem;
  const v2f* SC2 = (const v2f*)stf;
  const v2f* SH2 = (const v2f*)(stf + Dd);

  v8f acc[8];
  #pragma unroll
  for (int t = 0; t < 8; ++t) acc[t] = (v8f){0.f, 0.f, 0.f, 0.f, 0.f, 0.f, 0.f, 0.f};

  #pragma unroll 4
  for (int kk = 0; kk < 32; ++kk) {
    const int kb = kk * 4 + half * 2;         // this lane's K pair per ISA 16x4 f32 A layout
    v2f a = Arow2[kb >> 1];
    if (DO_TF) {                              // fused BN + ReLU on the input activations
      v2f s = SC2[kb >> 1];
      v2f h = SH2[kb >> 1];
      a.x = fmaxf(fmaf(a.x, s.x, h.x), 0.f);
      a.y = fmaxf(fmaf(a.y, s.y, h.y), 0.f);
    }
    #pragma unroll
    for (int t = 0; t < 8; ++t) {
      v2f b = B2[(t * 16 + l16) * BSTR2 + (kb >> 1)];
      acc[t] = __builtin_amdgcn_wmma_f32_16x16x4_f32(
          false, a, false, b, (short)0, acc[t], false, false);
    }
  }

  #pragma unroll
  for (int t = 0; t < 8; ++t) {
    const int col = t * 16 + l16;
    const float bv = HAS_BIAS ? bias[col] : 0.f;
    float ls = 0.f, lsq = 0.f;
    #pragma unroll
    for (int i = 0; i < 8; ++i) {
      int row = wrow + half * 8 + i;          // C/D layout: vgpr i, halves hold M=i / M=8+i
      float v = acc[t][i] + bv;
      if (RELU_OUT) v = fmaxf(v, 0.f);
      if (row < NNODES) {
        Y[(size_t)row * Dd + col] = v;
        ls += v;
        lsq += v * v;
      }
    }
    if (DO_STATS) {
      atomicAdd(&sstat[col], ls);
      atomicAdd(&sstat[Dd + col], lsq);
    }
  }
  if (DO_STATS) {
    __syncthreads();
    if (tid < Dd) {
      unsafeAtomicAdd(&gsum[tid], sstat[tid]);
      unsafeAtomicAdd(&gsq[tid], sstat[Dd + tid]);
    }
  }
}

// ---------------- BN finalize: per-column scale/shift from sums ---------------------------
__global__ void bnfin_kernel(const float* __restrict__ gsum, const float* __restrict__ gsq,
                             const float* __restrict__ gamma, const float* __restrict__ beta,
                             float* __restrict__ scale, float* __restrict__ shift) {
  int d = threadIdx.x;
  if (d < Dd) {
    float mean = gsum[d] * (1.f / NNODES);
    float var  = gsq[d] * (1.f / NNODES) - mean * mean;
    float inv  = rsqrtf(var + 1e-5f);
    float sc   = gamma[d] * inv;
    scale[d] = sc;
    shift[d] = beta[d] - mean * sc;
  }
}

extern "C" void kernel_launch(void* const* d_in, const int* in_sizes, int n_in,
                              void* d_out, int out_size, void* d_ws, size_t ws_size,
                              hipStream_t stream) {
  (void)in_sizes; (void)n_in; (void)out_size; (void)ws_size;
  const float*     x   = (const float*)d_in[0];
  const long long* ei  = (const long long*)d_in[1];   // int64 (2, NEDGES)
  const float*     W1  = (const float*)d_in[2];
  const float*     b1  = (const float*)d_in[3];
  const float*     g1  = (const float*)d_in[4];
  const float*     be1 = (const float*)d_in[5];
  const float*     W2  = (const float*)d_in[6];
  const float*     b2  = (const float*)d_in[7];
  const float*     g2  = (const float*)d_in[8];
  const float*     be2 = (const float*)d_in[9];
  const float*     W3  = (const float*)d_in[10];
  float* out = (float*)d_out;

  float* H     = (float*)d_ws;                        // 100000*128 floats (51.2 MB)
  float* stats = H + (size_t)NNODES * Dd;             // 1024 floats
  float* gsum1 = stats + 0,   *gsq1 = stats + 128;
  float* gsum2 = stats + 256, *gsq2 = stats + 384;
  float* sc1   = stats + 512, *sh1  = stats + 640;
  float* sc2   = stats + 768, *sh2  = stats + 896;

  const size_t shmem = (size_t)Dd * (2 * BSTR2) * sizeof(float);  // 66,560 B

  int n4 = NNODES * Dd / 4;
  init_kernel<<<(n4 + 255) / 256, 256, 0, stream>>>((const float4*)x, (float4*)H, stats);
  scatter_kernel<<<(NEDGES * 32) / 256, 256, 0, stream>>>((const float4*)x, ei, H);
  // y1 = H @ W1^T + b1 (in place over H; per-wave read-before-write makes aliasing safe)
  gemm_bn_kernel<false, true, true, false><<<NBLK, 256, shmem, stream>>>(
      H, W1, b1, nullptr, nullptr, H, gsum1, gsq1);
  bnfin_kernel<<<1, 128, 0, stream>>>(gsum1, gsq1, g1, be1, sc1, sh1);
  // y2 = relu(bn1(y1)) @ W2^T + b2  -> stored in d_out
  gemm_bn_kernel<true, true, true, false><<<NBLK, 256, shmem, stream>>>(
      H, W2, b2, sc1, sh1, out, gsum2, gsq2);
  bnfin_kernel<<<1, 128, 0, stream>>>(gsum2, gsq2, g2, be2, sc2, sh2);
  // out = relu( relu(bn2(y2)) @ W3^T )  (in place over d_out)
  gemm_bn_kernel<true, false, false, true><<<NBLK, 256, shmem, stream>>>(
      out, W3, nullptr, sc2, sh2, out, nullptr, nullptr);
}